// DeformableConv2d_75471165325937
// MI455X (gfx1250) — compile-verified
//
#include <hip/hip_runtime.h>
#include <hip/hip_bf16.h>

typedef __attribute__((ext_vector_type(2))) float v2f;
typedef __attribute__((ext_vector_type(8))) float v8f;

#define BATCH 4
#define CH    64
#define HH    128
#define WW    128
#define GRP   8
#define CG    8
#define KTAP  9
#define OFFC  144            // 2*G*K
#define KD    576            // CH * KTAP
#define TILE  32             // pixels per block (two 16-wide WMMA N tiles)
#define NTW   (WW / TILE)    // 4
#define PSTR  40             // padded panel row stride (floats): 2*40 mod 64 = 16
                             //  -> sel=0/sel=1 half-waves hit disjoint bank windows

// ---------------------------------------------------------------------------
// Kernel 1: offset-field conv (3x3, Cin=64, Cout=144) as implicit GEMM.
//   M=144 (9 wave tiles), N=32 (2 WMMA tiles/wave), K=576.
//   halo patch -> im2col panel (both LDS), then pure WMMA loop.
// ---------------------------------------------------------------------------
__global__ __launch_bounds__(288)
void offconv_wmma_kernel(const float* __restrict__ feat,
                         const float* __restrict__ off_w,
                         const float* __restrict__ off_b,
                         float* __restrict__ offset_out)
{
    __shared__ float patch[CH * 3 * 34];   // 26.1 KB: 64ch x 3 rows x (32+2) cols
    __shared__ float panel[KD * PSTR];     // 92.2 KB im2col B panel

    const int blk = blockIdx.x;
    const int tw  = blk & (NTW - 1);
    const int h   = (blk >> 2) & (HH - 1);
    const int b   = blk >> 9;
    const int w0  = tw * TILE;
    const int tid = threadIdx.x;

    const float* fb = feat + (size_t)b * CH * HH * WW;

    // --- stage halo patch (zero-padded) -----------------------------------
    for (int i = tid; i < CH * 3 * 34; i += 288) {
        int c   = i / 102;
        int r   = i - c * 102;
        int kh  = r / 34;
        int col = r - kh * 34;
        int hy  = h + kh - 1;
        int wx  = w0 + col - 1;
        float v = 0.0f;
        if (hy >= 0 && hy < HH && wx >= 0 && wx < WW)
            v = fb[(c * HH + hy) * WW + wx];
        patch[i] = v;
    }
    __syncthreads();

    // --- expand to im2col panel: panel[kd][n] = patch[c][kh][n+kw] --------
    for (int i = tid; i < KD * TILE; i += 288) {   // 18432/288 = 64 each
        int kd = i >> 5;
        int n  = i & 31;
        int c  = kd / 9;
        int t  = kd - c * 9;
        int kh = t / 3;
        int kw = t - kh * 3;
        panel[kd * PSTR + n] = patch[c * 102 + kh * 34 + n + kw];
    }
    __syncthreads();

    // --- fp32 WMMA GEMM ---------------------------------------------------
    const int wave = tid >> 5;          // 0..8 -> M tile
    const int lane = tid & 31;
    const int m0   = wave * 16;
    const int sel  = lane >> 4;
    const int nm   = lane & 15;

    const float* wrow = off_w + (size_t)(m0 + nm) * KD + 2 * sel;

    v8f acc0 = {};
    v8f acc1 = {};
    #pragma unroll 4
    for (int kk = 0; kk < KD; kk += 4) {
        v2f a = *(const v2f*)(wrow + kk);   // 8B-aligned -> global_load_b64

        const int r0 = (kk + 2 * sel) * PSTR;
        v2f b0, b1;
        b0.x = panel[r0 + nm];
        b0.y = panel[r0 + PSTR + nm];
        b1.x = panel[r0 + 16 + nm];
        b1.y = panel[r0 + PSTR + 16 + nm];

        acc0 = __builtin_amdgcn_wmma_f32_16x16x4_f32(
                   false, a, false, b0, (short)0, acc0, false, false);
        acc1 = __builtin_amdgcn_wmma_f32_16x16x4_f32(
                   false, a, false, b1, (short)0, acc1, false, false);
    }

    // --- store ------------------------------------------------------------
    for (int r = 0; r < 8; ++r) {
        int m = m0 + r + 8 * sel;
        float bb = off_b[m];
        size_t base = (((size_t)b * OFFC + m) * HH + h) * WW + w0;
        offset_out[base + nm]      = acc0[r] + bb;
        offset_out[base + 16 + nm] = acc1[r] + bb;
    }
}

// ---------------------------------------------------------------------------
// Kernel 2: fused bilinear sampling + deformable GEMM.
//   Phase 1 (256 threads): 576x32 sampled panel in LDS; bilinear weights
//   computed once per (g,k,pixel), reused over the 8 channels of the group;
//   gathers hit the 192MB L2 (feat is 16.8MB).
//   Phase 2: 8 waves = 4 M-tiles x 2 N-tiles, fp32 WMMA over K=576.
// ---------------------------------------------------------------------------
__global__ __launch_bounds__(256)
void deform_wmma_kernel(const float* __restrict__ feat,
                        const float* __restrict__ weight,
                        const float* __restrict__ bias,
                        const float* __restrict__ offset,
                        float* __restrict__ out)
{
    __shared__ float panel[KD * PSTR];  // 92.2 KB sampled B panel

    const int blk = blockIdx.x;
    const int tw  = blk & (NTW - 1);
    const int h   = (blk >> 2) & (HH - 1);
    const int b   = blk >> 9;
    const int w0  = tw * TILE;
    const int tid = threadIdx.x;

    const float* fb = feat   + (size_t)b * CH * HH * WW;
    const float* ob = offset + (size_t)b * OFFC * HH * WW;

    // --- Phase 1: sampling (72 (g,k) x 32 px; 9 iters/thread) -----------
    for (int i = tid; i < GRP * KTAP * TILE; i += 256) {
        int gk  = i >> 5;               // 0..71
        int pix = i & 31;
        int g   = gk / KTAP;
        int k   = gk - g * KTAP;
        int w   = w0 + pix;

        float dy = ob[(size_t)((2 * gk + 0) * HH + h) * WW + w];
        float dx = ob[(size_t)((2 * gk + 1) * HH + h) * WW + w];

        float y = (float)h + (float)(k / 3 - 1) + dy;
        float x = (float)w + (float)(k % 3 - 1) + dx;

        float y0f = floorf(y), x0f = floorf(x);
        float wy = y - y0f, wx = x - x0f;
        int y0 = (int)y0f, x0 = (int)x0f;
        int y1 = y0 + 1,   x1 = x0 + 1;

        float my0 = (y0 >= 0 && y0 <= HH - 1) ? 1.0f : 0.0f;
        float my1 = (y1 >= 0 && y1 <= HH - 1) ? 1.0f : 0.0f;
        float mx0 = (x0 >= 0 && x0 <= WW - 1) ? 1.0f : 0.0f;
        float mx1 = (x1 >= 0 && x1 <= WW - 1) ? 1.0f : 0.0f;

        int y0c = min(max(y0, 0), HH - 1);
        int y1c = min(max(y1, 0), HH - 1);
        int x0c = min(max(x0, 0), WW - 1);
        int x1c = min(max(x1, 0), WW - 1);

        float w00 = (1.0f - wy) * (1.0f - wx) * my0 * mx0;
        float w01 = (1.0f - wy) * wx          * my0 * mx1;
        float w10 = wy          * (1.0f - wx) * my1 * mx0;
        float w11 = wy          * wx          * my1 * mx1;

        int i00 = y0c * WW + x0c;
        int i01 = y0c * WW + x1c;
        int i10 = y1c * WW + x0c;
        int i11 = y1c * WW + x1c;

        const float* chp = fb + (size_t)(g * CG) * HH * WW;
        #pragma unroll
        for (int c = 0; c < CG; ++c) {
            float v = w00 * chp[i00] + w01 * chp[i01]
                    + w10 * chp[i10] + w11 * chp[i11];
            panel[((g * CG + c) * KTAP + k) * PSTR + pix] = v;
            chp += HH * WW;
        }
    }
    __syncthreads();

    // --- Phase 2: fp32 WMMA GEMM, 8 waves = 4 M-tiles x 2 N-tiles ---------
    const int wave = tid >> 5;
    const int lane = tid & 31;
    const int mt   = wave & 3;          // M tile
    const int nt   = wave >> 2;         // N tile
    const int m0   = mt * 16;
    const int n0   = nt * 16;
    const int sel  = lane >> 4;
    const int nm   = lane & 15;

    const float* wrow = weight + (size_t)(m0 + nm) * KD + 2 * sel;

    v8f acc = {};
    #pragma unroll 4
    for (int kk = 0; kk < KD; kk += 4) {
        v2f a = *(const v2f*)(wrow + kk);   // 8B-aligned -> global_load_b64

        const int r0 = (kk + 2 * sel) * PSTR + n0;
        v2f bf;
        bf.x = panel[r0 + nm];
        bf.y = panel[r0 + PSTR + nm];

        acc = __builtin_amdgcn_wmma_f32_16x16x4_f32(
                  false, a, false, bf, (short)0, acc, false, false);
    }

    for (int r = 0; r < 8; ++r) {
        int m = m0 + r + 8 * sel;
        out[(((size_t)b * CH + m) * HH + h) * WW + w0 + n0 + nm] = acc[r] + bias[m];
    }
}

// ---------------------------------------------------------------------------
extern "C" void kernel_launch(void* const* d_in, const int* in_sizes, int n_in,
                              void* d_out, int out_size, void* d_ws, size_t ws_size,
                              hipStream_t stream) {
    (void)in_sizes; (void)n_in; (void)out_size; (void)ws_size;

    const float* feat   = (const float*)d_in[0];
    const float* weight = (const float*)d_in[1];
    const float* bias   = (const float*)d_in[2];
    const float* off_w  = (const float*)d_in[3];
    const float* off_b  = (const float*)d_in[4];
    float* out    = (float*)d_out;
    float* offmap = (float*)d_ws;  // B*144*H*W fp32 = 37.75 MB scratch

    const int tiles = BATCH * HH * NTW; // 2048

    offconv_wmma_kernel<<<tiles, 288, 0, stream>>>(feat, off_w, off_b, offmap);
    deform_wmma_kernel <<<tiles, 256, 0, stream>>>(feat, weight, bias, offmap, out);
}